// BASE_49821620633700
// MI455X (gfx1250) — compile-verified
//
#include <hip/hip_runtime.h>
#include <math.h>

typedef float v2f __attribute__((ext_vector_type(2)));
typedef float v8f __attribute__((ext_vector_type(8)));

#define HWN 1024   // 32*32 positions
#define CCH 512    // channels
#define NSZ 32

// ---- fp32 WMMA 16x16x4 (D = A(16x4) * B(4x16) + C), wave32 ----
__device__ __forceinline__ v8f wmma4(v2f a, v2f b, v8f c) {
    return __builtin_amdgcn_wmma_f32_16x16x4_f32(
        /*neg_a=*/false, a, /*neg_b=*/false, b,
        /*c_mod=*/(short)0, c, /*reuse_a=*/false, /*reuse_b=*/false);
}

// ---------------- normalized separable gaussian weights Gn[i][x] ----------------
__global__ void k_gauss(float* __restrict__ G) {  // <<<1, 1024>>>
    __shared__ float e[1024];
    __shared__ float S[32];
    int t = threadIdx.x;
    int i = t >> 5, x = t & 31;
    float d = (float)(i - x);
    float v = __expf(-d * d * (1.0f / 4.5f));   // 2*v^2 = 4.5, v=1.5
    e[t] = v;
    __syncthreads();
    if (t < 32) { float s = 0.f; for (int k = 0; k < 32; ++k) s += e[t * 32 + k]; S[t] = s; }
    __syncthreads();
    G[t] = v / S[i];   // row-normalized; 2D normalization factorizes
}

// ---------------- per-channel global average ----------------
__global__ void k_mean(const float* __restrict__ x, float* __restrict__ meanv) { // <<<512,256>>>
    __shared__ float red[256];
    int c = blockIdx.x, t = threadIdx.x;
    float s = 0.f;
    for (int i = t; i < HWN; i += 256) s += x[c * HWN + i];
    red[t] = s; __syncthreads();
    for (int off = 128; off > 0; off >>= 1) { if (t < off) red[t] += red[t + off]; __syncthreads(); }
    if (t == 0) meanv[c] = red[0] * (1.0f / HWN);
}

// ---------------- SE: FC(512->32)+ReLU, FC(32->512)+Sigmoid ----------------
__global__ void k_se(const float* __restrict__ meanv,
                     const float* __restrict__ w1, const float* __restrict__ b1,
                     const float* __restrict__ w2, const float* __restrict__ b2,
                     float* __restrict__ yscale) {  // <<<1, 512>>>
    __shared__ float sm[CCH];
    __shared__ float sh[32];
    int t = threadIdx.x;
    sm[t] = meanv[t];
    __syncthreads();
    if (t < 32) {
        float a = b1[t];
        for (int c = 0; c < CCH; ++c) a += w1[t * CCH + c] * sm[c];
        sh[t] = fmaxf(a, 0.f);
    }
    __syncthreads();
    float a = b2[t];
    for (int r = 0; r < 32; ++r) a += w2[t * 32 + r] * sh[r];
    yscale[t] = 1.f / (1.f + __expf(-a));
}

// ---------------- out32 = x*y ; sigmoid ; produce [c][pos] and [pos][c] layouts ----------------
__global__ void k_scale(const float* __restrict__ x, const float* __restrict__ yscale,
                        float* __restrict__ out_cp, float* __restrict__ out_pc,
                        float* __restrict__ sig_pc) {  // <<<512,256>>>
    int c = blockIdx.x, t = threadIdx.x;
    float s = yscale[c];
    for (int pos = t; pos < HWN; pos += 256) {
        float v = x[c * HWN + pos] * s;
        out_cp[c * HWN + pos] = v;
        out_pc[pos * CCH + c] = v;
        sig_pc[pos * CCH + c] = 1.f / (1.f + __expf(-v));
    }
}

// ---------------- gaussian blur pass 1: Tmp_c = Gn(32x32) x M_c(32x32), one wave per channel ----
__global__ void k_blur1(const float* __restrict__ G, const float* __restrict__ out_cp,
                        float* __restrict__ gtmp) {  // <<<64,256>>> -> 512 waves
    int wave = (blockIdx.x * blockDim.x + threadIdx.x) >> 5;
    int lane = threadIdx.x & 31;
    int half = lane >> 4, l = lane & 15;
    const float* M = out_cp + wave * HWN;
    v8f acc[2][2] = {};
    for (int k0 = 0; k0 < 32; k0 += 4) {
        int ka = k0 + 2 * half;
        v2f a0, a1, b0, b1;
        a0.x = G[l * 32 + ka];            a0.y = G[l * 32 + ka + 1];
        a1.x = G[(16 + l) * 32 + ka];     a1.y = G[(16 + l) * 32 + ka + 1];
        b0.x = M[ka * 32 + l];            b0.y = M[(ka + 1) * 32 + l];
        b1.x = M[ka * 32 + 16 + l];       b1.y = M[(ka + 1) * 32 + 16 + l];
        acc[0][0] = wmma4(a0, b0, acc[0][0]);
        acc[0][1] = wmma4(a0, b1, acc[0][1]);
        acc[1][0] = wmma4(a1, b0, acc[1][0]);
        acc[1][1] = wmma4(a1, b1, acc[1][1]);
    }
    float* T = gtmp + wave * HWN;
    for (int mt = 0; mt < 2; ++mt)
        for (int nt = 0; nt < 2; ++nt)
            for (int r = 0; r < 8; ++r)
                T[(mt * 16 + half * 8 + r) * 32 + nt * 16 + l] = acc[mt][nt][r];
}

// ---------------- gaussian blur pass 2: Out_c = Tmp_c x Gn^T ; scatter to (p,c)-flat --------
__global__ void k_blur2(const float* __restrict__ G, const float* __restrict__ gtmp,
                        float* __restrict__ gusflat) {  // <<<64,256>>> -> 512 waves
    int wave = (blockIdx.x * blockDim.x + threadIdx.x) >> 5;
    int lane = threadIdx.x & 31;
    int half = lane >> 4, l = lane & 15;
    const float* T = gtmp + wave * HWN;
    v8f acc[2][2] = {};
    for (int k0 = 0; k0 < 32; k0 += 4) {
        int ka = k0 + 2 * half;
        v2f a0, a1, b0, b1;
        a0.x = T[l * 32 + ka];            a0.y = T[l * 32 + ka + 1];
        a1.x = T[(16 + l) * 32 + ka];     a1.y = T[(16 + l) * 32 + ka + 1];
        // B[y][j] = Gn[j][y]
        b0.x = G[l * 32 + ka];            b0.y = G[l * 32 + ka + 1];
        b1.x = G[(16 + l) * 32 + ka];     b1.y = G[(16 + l) * 32 + ka + 1];
        acc[0][0] = wmma4(a0, b0, acc[0][0]);
        acc[0][1] = wmma4(a0, b1, acc[0][1]);
        acc[1][0] = wmma4(a1, b0, acc[1][0]);
        acc[1][1] = wmma4(a1, b1, acc[1][1]);
    }
    for (int mt = 0; mt < 2; ++mt)
        for (int nt = 0; nt < 2; ++nt)
            for (int r = 0; r < 8; ++r) {
                int p = (mt * 16 + half * 8 + r) * 32 + nt * 16 + l;  // i*32 + j
                gusflat[p * CCH + wave] = acc[mt][nt][r];
            }
}

// ---------------- 3x3 patch attention: dots -> softmax -> weighted sum ----------------
__global__ void k_patch(const float* __restrict__ out_pc, const float* __restrict__ sig_pc,
                        float* __restrict__ csaflat) {  // <<<1024,256>>>
    __shared__ float s_dot[9];
    __shared__ float s_w[9];
    int p = blockIdx.x, t = threadIdx.x;
    int py = p >> 5, px = p & 31;
    if (t < 9) s_dot[t] = 0.f;
    __syncthreads();
    float c0 = sig_pc[p * CCH + t];
    float c1 = sig_pc[p * CCH + t + 256];
    for (int ij = 0; ij < 9; ++ij) {
        int ny = py + ij / 3 - 1, nx = px + ij % 3 - 1;
        float partial = 0.f;
        if (ny >= 0 && ny < 32 && nx >= 0 && nx < 32) {
            int np = ny * 32 + nx;
            partial = c0 * sig_pc[np * CCH + t] + c1 * sig_pc[np * CCH + t + 256];
        }
        for (int off = 16; off > 0; off >>= 1) partial += __shfl_down(partial, off);
        if ((t & 31) == 0) atomicAdd(&s_dot[ij], partial);   // ds_add_f32
    }
    __syncthreads();
    if (t == 0) {
        float mx = -1e30f;
        for (int ij = 0; ij < 9; ++ij) { float a = s_dot[ij] * (1.f / (float)CCH); s_w[ij] = a; mx = fmaxf(mx, a); }
        float den = 0.f;
        for (int ij = 0; ij < 9; ++ij) { float e = __expf(s_w[ij] - mx); s_w[ij] = e; den += e; }
        float inv = 1.f / den;
        for (int ij = 0; ij < 9; ++ij) s_w[ij] *= inv;
    }
    __syncthreads();
    float o0 = 0.f, o1 = 0.f;
    for (int ij = 0; ij < 9; ++ij) {
        int ny = py + ij / 3 - 1, nx = px + ij % 3 - 1;
        if (ny >= 0 && ny < 32 && nx >= 0 && nx < 32) {
            int np = ny * 32 + nx;
            float w = s_w[ij];
            o0 += w * out_pc[np * CCH + t];
            o1 += w * out_pc[np * CCH + t + 256];
        }
    }
    csaflat[p * CCH + t] = o0;
    csaflat[p * CCH + t + 256] = o1;
}

// ---------------- down 1x1 conv: O(512x1024) = W(512x1024) x [gus;csa](1024x1024) ----------
// The torch .view flat-reinterpretation means the (p,c)-major flat buffers read
// directly as row-major (512,1024) B-halves.
__global__ void k_down(const float* __restrict__ Wt, const float* __restrict__ gus,
                       const float* __restrict__ csa, float* __restrict__ opre) { // <<<64,256>>>
    int wave = (blockIdx.x * blockDim.x + threadIdx.x) >> 5;   // 512 waves
    int lane = threadIdx.x & 31;
    int half = lane >> 4, l = lane & 15;
    int mt = wave & 31;    // 32 M tiles of 16
    int ns = wave >> 5;    // 16 N strips of 64
    int m0 = mt * 16, n0 = ns * 64;
    v8f acc[4] = {};
    const float* Arow = Wt + (m0 + l) * 1024;
    for (int phase = 0; phase < 2; ++phase) {
        const float* B = phase ? csa : gus;
        const float* Ap = Arow + phase * 512;
        for (int k0 = 0; k0 < 512; k0 += 4) {
            int ka = k0 + 2 * half;
            if (k0 + 16 < 512) __builtin_prefetch(&B[(k0 + 16) * 1024 + n0 + l], 0, 1);
            v2f a; a.x = Ap[ka]; a.y = Ap[ka + 1];
#pragma unroll
            for (int nt = 0; nt < 4; ++nt) {
                v2f b;
                b.x = B[ka * 1024 + n0 + nt * 16 + l];
                b.y = B[(ka + 1) * 1024 + n0 + nt * 16 + l];
                acc[nt] = wmma4(a, b, acc[nt]);
            }
        }
    }
    for (int nt = 0; nt < 4; ++nt)
        for (int r = 0; r < 8; ++r)
            opre[(m0 + half * 8 + r) * 1024 + n0 + nt * 16 + l] = acc[nt][r];
}

// ---------------- InstanceNorm (population var) + LeakyReLU(0.2) ----------------
__global__ void k_norm(const float* __restrict__ opre, float* __restrict__ out) { // <<<512,256>>>
    __shared__ float rs[256], rq[256];
    int c = blockIdx.x, t = threadIdx.x;
    float s = 0.f, q = 0.f;
    for (int i = t; i < HWN; i += 256) { float v = opre[c * HWN + i]; s += v; q += v * v; }
    rs[t] = s; rq[t] = q; __syncthreads();
    for (int off = 128; off > 0; off >>= 1) {
        if (t < off) { rs[t] += rs[t + off]; rq[t] += rq[t + off]; }
        __syncthreads();
    }
    float m = rs[0] * (1.0f / HWN);
    float var = rq[0] * (1.0f / HWN) - m * m;
    float inv = rsqrtf(var + 1e-5f);
    for (int i = t; i < HWN; i += 256) {
        float v = (opre[c * HWN + i] - m) * inv;
        out[c * HWN + i] = v >= 0.f ? v : 0.2f * v;
    }
}

extern "C" void kernel_launch(void* const* d_in, const int* in_sizes, int n_in,
                              void* d_out, int out_size, void* d_ws, size_t ws_size,
                              hipStream_t stream) {
    const float* x     = (const float*)d_in[0];   // (1,512,32,32)
    const float* se_w1 = (const float*)d_in[1];   // (32,512)
    const float* se_b1 = (const float*)d_in[2];   // (32)
    const float* se_w2 = (const float*)d_in[3];   // (512,32)
    const float* se_b2 = (const float*)d_in[4];   // (512)
    const float* dw    = (const float*)d_in[5];   // (512,1024)
    float* out = (float*)d_out;                   // (1,512,32,32)

    float* ws = (float*)d_ws;
    float* G       = ws;                 // 1024
    float* meanv   = G + 1024;           // 512
    float* yscale  = meanv + 512;        // 512
    float* out_cp  = yscale + 512;       // 512*1024
    float* out_pc  = out_cp + CCH * HWN; // 512*1024
    float* sig_pc  = out_pc + CCH * HWN; // 512*1024
    float* gtmp    = sig_pc + CCH * HWN; // 512*1024
    float* gusflat = gtmp   + CCH * HWN; // 1024*512 (p,c)
    float* csaflat = gusflat+ HWN * CCH; // 1024*512 (p,c)
    float* opre    = csaflat+ HWN * CCH; // 512*1024

    k_gauss<<<1, 1024, 0, stream>>>(G);
    k_mean <<<512, 256, 0, stream>>>(x, meanv);
    k_se   <<<1, 512, 0, stream>>>(meanv, se_w1, se_b1, se_w2, se_b2, yscale);
    k_scale<<<512, 256, 0, stream>>>(x, yscale, out_cp, out_pc, sig_pc);
    k_blur1<<<64, 256, 0, stream>>>(G, out_cp, gtmp);
    k_blur2<<<64, 256, 0, stream>>>(G, gtmp, gusflat);
    k_patch<<<1024, 256, 0, stream>>>(out_pc, sig_pc, csaflat);
    k_down <<<64, 256, 0, stream>>>(dw, gusflat, csaflat, opre);
    k_norm <<<512, 256, 0, stream>>>(opre, out);
}